// GraphConvGenAggWrapper_57535381897288
// MI455X (gfx1250) — compile-verified
//
#include <hip/hip_runtime.h>

#define N_NODES 100000
#define N_EDGES 1000000
#define D       64

typedef __attribute__((ext_vector_type(2))) float v2f;
typedef __attribute__((ext_vector_type(4))) float v4f;
typedef __attribute__((ext_vector_type(8))) float v8f;

// ---- Native CDNA5 f32 atomics (guaranteed, no CAS-loop fallback) ----------
__device__ __forceinline__ void atomic_max_f32(float* p, float v) {
    asm volatile("global_atomic_max_num_f32 %0, %1, off scope:SCOPE_DEV"
                 :: "v"(p), "v"(v) : "memory");
}
__device__ __forceinline__ void atomic_add_f32(float* p, float v) {
    asm volatile("global_atomic_add_f32 %0, %1, off scope:SCOPE_DEV"
                 :: "v"(p), "v"(v) : "memory");
}

// ---- Kernel 0: init workspace (m = -inf, denom = num = 0), float4 --------
__global__ void init_ws_kernel(float* __restrict__ m,
                               float* __restrict__ den,
                               float* __restrict__ num) {
    int i = (blockIdx.x * blockDim.x + threadIdx.x) * 4;
    if (i < N_NODES * D) {
        const float ninf = -__builtin_inff();
        *(v4f*)(m + i)   = (v4f){ninf, ninf, ninf, ninf};
        *(v4f*)(den + i) = (v4f){0.f, 0.f, 0.f, 0.f};
        *(v4f*)(num + i) = (v4f){0.f, 0.f, 0.f, 0.f};
    }
}

// ---- Kernel 1: per-(dst,channel) segmented max of t*x[src] ----------------
// One wave per edge: lane L covers channels 2L, 2L+1 (float2 coalesced).
// Edge id forced wave-uniform so src/dst/t become scalar s_loads.
__global__ void edge_max_kernel(const float* __restrict__ x,
                                const long long* __restrict__ ei,
                                float* __restrict__ m,
                                const float* __restrict__ tp) {
    int wave = __builtin_amdgcn_readfirstlane(threadIdx.x >> 5);
    int e    = blockIdx.x * 8 + wave;               // uniform per wave
    int lane = threadIdx.x & 31;
    if (e >= N_EDGES) return;
    float t = *tp;                                   // scalar load
    long long src = ei[e];                           // scalar load
    long long dst = ei[N_EDGES + e];                 // scalar load
    int c = lane * 2;
    v2f xj = *(const v2f*)(x + src * D + c);
    float* mp = m + dst * D + c;
    atomic_max_f32(mp,     t * xj.x);
    atomic_max_f32(mp + 1, t * xj.y);
}

// ---- Kernel 2: w = exp(t*xj - m[dst]); denom += w; num += w*xj ------------
__global__ void edge_sum_kernel(const float* __restrict__ x,
                                const long long* __restrict__ ei,
                                const float* __restrict__ m,
                                float* __restrict__ den,
                                float* __restrict__ num,
                                const float* __restrict__ tp) {
    int wave = __builtin_amdgcn_readfirstlane(threadIdx.x >> 5);
    int e    = blockIdx.x * 8 + wave;               // uniform per wave
    int lane = threadIdx.x & 31;
    if (e >= N_EDGES) return;
    float t = *tp;
    long long src = ei[e];
    long long dst = ei[N_EDGES + e];
    int c = lane * 2;
    v2f xj = *(const v2f*)(x + src * D + c);
    v2f mm = *(const v2f*)(m + dst * D + c);
    float w0 = __expf(t * xj.x - mm.x);
    float w1 = __expf(t * xj.y - mm.y);
    float* dp = den + dst * D + c;
    float* np = num + dst * D + c;
    atomic_add_f32(dp,     w0);
    atomic_add_f32(dp + 1, w1);
    atomic_add_f32(np,     w0 * xj.x);
    atomic_add_f32(np + 1, w1 * xj.y);
}

// ---- Kernel 3: agg = where(den>0, num/max(den,1e-16), 0) ------------------
// Precise division happens ONCE per element here (memory-bound pass), keeping
// the WMMA loop clean. Writes over the dead `m` buffer.
__global__ void agg_kernel(const float* __restrict__ num,
                           const float* __restrict__ den,
                           float* __restrict__ agg) {
    int i = blockIdx.x * blockDim.x + threadIdx.x;
    if (i < N_NODES * D) {
        float de = den[i];
        float nu = num[i];
        agg[i] = de > 0.0f ? nu / fmaxf(de, 1e-16f) : 0.0f;
    }
}

// ---- Kernel 4: out = agg @ Wrel^T + x @ Wroot^T + bias via WMMA f32 -------
// Wave-per-16x16-output-tile. A-layout (16x4 f32): lane holds row M=lane%16,
// VGPR0/1 = K+(lane>=16?2:0)+{0,1}. B-layout (4x16): N = lane%16, same K split.
// C/D: vgpr i -> M = i + (lane>=16 ? 8 : 0), N = lane%16.
__global__ void __launch_bounds__(256)
node_gemm_kernel(const float* __restrict__ x,
                 const float* __restrict__ agg,
                 const float* __restrict__ Wrel,
                 const float* __restrict__ Wroot,
                 const float* __restrict__ bias,
                 float* __restrict__ out) {
    int wave = __builtin_amdgcn_readfirstlane(threadIdx.x >> 5);
    int lane = threadIdx.x & 31;
    int tile = blockIdx.x * 8 + wave;        // 25000 tiles = 6250 (M) x 4 (N)
    int tm   = tile >> 2;
    int tn   = tile & 3;

    int row = tm * 16 + (lane & 15);         // A row (node)
    int col = tn * 16 + (lane & 15);         // B col (output channel)
    int kq  = (lane >> 4) << 1;              // K sub-offset: 0 or 2

    const float* aggr   = agg   + (size_t)row * D + kq;
    const float* xr     = x     + (size_t)row * D + kq;
    const float* wrelr  = Wrel  + (size_t)col * D + kq;
    const float* wrootr = Wroot + (size_t)col * D + kq;

    v8f acc = {};
#pragma unroll
    for (int k0 = 0; k0 < D; k0 += 4) {
        v2f a1 = *(const v2f*)(aggr + k0);
        v2f b1 = *(const v2f*)(wrelr + k0);
        v2f a2 = *(const v2f*)(xr + k0);
        v2f b2 = *(const v2f*)(wrootr + k0);
        acc = __builtin_amdgcn_wmma_f32_16x16x4_f32(
                  false, a1, false, b1, (short)0, acc, false, false);
        acc = __builtin_amdgcn_wmma_f32_16x16x4_f32(
                  false, a2, false, b2, (short)0, acc, false, false);
    }

    float bv = bias[col];
    int mrow0 = tm * 16 + ((lane >> 4) << 3);
#pragma unroll
    for (int i = 0; i < 8; ++i)
        out[(size_t)(mrow0 + i) * D + col] = acc[i] + bv;
}

extern "C" void kernel_launch(void* const* d_in, const int* in_sizes, int n_in,
                              void* d_out, int out_size, void* d_ws, size_t ws_size,
                              hipStream_t stream) {
    const float*     x     = (const float*)d_in[0];
    const long long* ei    = (const long long*)d_in[1];   // int64 [2, E]
    const float*     Wrel  = (const float*)d_in[2];
    const float*     Wroot = (const float*)d_in[3];
    const float*     bias  = (const float*)d_in[4];
    const float*     t     = (const float*)d_in[5];
    float*           out   = (float*)d_out;

    float* m   = (float*)d_ws;                  // [N, D] max, later reused as agg
    float* den = m   + (size_t)N_NODES * D;     // [N, D]
    float* num = den + (size_t)N_NODES * D;     // [N, D]  total 76.8 MB

    init_ws_kernel<<<(N_NODES * D / 4 + 255) / 256, 256, 0, stream>>>(m, den, num);
    edge_max_kernel<<<(N_EDGES * 32) / 256, 256, 0, stream>>>(x, ei, m, t);
    edge_sum_kernel<<<(N_EDGES * 32) / 256, 256, 0, stream>>>(x, ei, m, den, num, t);
    agg_kernel<<<(N_NODES * D + 255) / 256, 256, 0, stream>>>(num, den, /*agg=*/m);
    node_gemm_kernel<<<(N_NODES / 16) * (D / 16) / 8, 256, 0, stream>>>(
        x, /*agg=*/m, Wrel, Wroot, bias, out);
}